// DilateLoss_34634616275447
// MI455X (gfx1250) — compile-verified
//
#include <hip/hip_runtime.h>
#include <hip/hip_bf16.h>

#define NSEQ   256
#define NB     64
#define STRIDE 258            // 258x258 R matrix, stride 258 -> conflict-free diagonals
#define BIGV   1e8f
#define NEGB  -1e30f
#define GAMMA  0.01f
#define INVG   100.0f
#define ALPHA_ 0.5f

typedef __attribute__((ext_vector_type(2))) float v2f;
typedef __attribute__((ext_vector_type(8))) float v8f;

// One workgroup (256 threads = 8 wave32) per batch element.
// Forward soft-DTW DP + backward gradient DP over anti-diagonals, all state in LDS.
__global__ void __launch_bounds__(256)
dilate_batch_kernel(const float* __restrict__ y_pred,   // o: columns of D
                    const float* __restrict__ y_true,   // t: rows of D
                    float* __restrict__ partials) {
    __shared__ float Rm[STRIDE * STRIDE];   // R[0..257][0..257], 266,256 B
    __shared__ float tl[272];               // t padded (tl[256..]=0)
    __shared__ float ol[272];               // o padded
    __shared__ float Eb[3][272];            // rolling E diagonals
    __shared__ float red[256];              // reduction scratch

    const int tid = threadIdx.x;
    const int b   = blockIdx.x;

    // ---- load inputs into LDS ----
    tl[tid] = y_true[b * NSEQ + tid];
    ol[tid] = y_pred[b * NSEQ + tid];
    if (tid < 16) { tl[256 + tid] = 0.0f; ol[256 + tid] = 0.0f; }

    // ---- R boundary init (phase 1: forward boundaries) ----
    for (int idx = tid; idx < STRIDE; idx += 256) {
        Rm[idx]          = (idx == 0) ? 0.0f : BIGV;   // row 0
        Rm[idx * STRIDE] = (idx == 0) ? 0.0f : BIGV;   // col 0
    }
    __syncthreads();
    // phase 2: backward sentinels (overwrite corners deterministically)
    for (int idx = tid; idx < STRIDE; idx += 256) {
        Rm[257 * STRIDE + idx] = NEGB;                 // row 257
        Rm[idx * STRIDE + 257] = NEGB;                 // col 257
    }
    __syncthreads();

    // ---- forward DP over anti-diagonals d = i + j ----
    for (int d = 2; d <= 2 * NSEQ; ++d) {
        const int lo = max(1, d - NSEQ);
        const int hi = min(NSEQ, d - 1);
        if (tid <= hi - lo) {
            const int i = lo + tid, j = d - i;
            const float a  = Rm[i * STRIDE + (j - 1)];       // V[i, j-1]
            const float bb = Rm[(i - 1) * STRIDE + (j - 1)]; // V[i-1, j-1]
            const float c  = Rm[(i - 1) * STRIDE + j];       // V[i-1, j]
            const float m  = fminf(a, fminf(bb, c));
            const float r  = __expf((m - a)  * INVG)
                           + __expf((m - bb) * INVG)
                           + __expf((m - c)  * INVG);
            const float df = tl[i - 1] - ol[j - 1];
            Rm[i * STRIDE + j] = df * df + m - GAMMA * __logf(r);
        }
        __syncthreads();
    }

    // ---- zero E buffers, seed E[256,256] = 1 ----
    for (int idx = tid; idx < 3 * 272; idx += 256) ((float*)Eb)[idx] = 0.0f;
    __syncthreads();
    if (tid == 0) Eb[(2 * NSEQ) % 3][NSEQ] = 1.0f;   // (512 % 3) == 2
    __syncthreads();

    // ---- backward DP: E[i,j] = wa*E[i+1,j] + wb*E[i,j+1] + wc*E[i+1,j+1] ----
    float acc = 0.0f;   // sum of E * (i-j)^2 (seed cell has omega = 0)
    for (int d = 2 * NSEQ - 1; d >= 2; --d) {
        const int cur = d % 3, n1 = (d + 1) % 3, n2 = (d + 2) % 3;
        const int lo = max(1, d - NSEQ);
        const int hi = min(NSEQ, d - 1);
        if (tid <= hi - lo) {
            const int i = lo + tid, j = d - i;
            const float rij  = Rm[i * STRIDE + j];
            const float tim1 = tl[i - 1], ti = tl[i];     // tl[256]=0 pad keeps this finite
            const float ojm1 = ol[j - 1], oj = ol[j];
            float dd = ti   - ojm1; dd *= dd;             // D[i+1, j]
            float dr = tim1 - oj;   dr *= dr;             // D[i, j+1]
            float dg = ti   - oj;   dg *= dg;             // D[i+1, j+1]
            // sentinel R = -1e30 makes out-of-grid weights underflow to exactly 0
            const float wa = __expf((Rm[(i + 1) * STRIDE + j]       - rij - dd) * INVG);
            const float wb = __expf((Rm[i * STRIDE + (j + 1)]       - rij - dr) * INVG);
            const float wc = __expf((Rm[(i + 1) * STRIDE + (j + 1)] - rij - dg) * INVG);
            const float e  = wa * Eb[n1][i + 1] + wb * Eb[n1][i] + wc * Eb[n2][i + 1];
            Eb[cur][i] = e;
            const int di = 2 * i - d;                     // i - j
            acc += e * (float)(di * di);
        }
        // kill stale values (from diagonal d+3) just below this diagonal's range
        if (tid < 2) {
            const int r0 = lo - 1 - tid;
            if (r0 >= 0) Eb[cur][r0] = 0.0f;
        }
        __syncthreads();
    }

    // ---- reduce temporal accumulator, emit combined per-batch partial ----
    red[tid] = acc;
    __syncthreads();
    for (int s = 128; s > 0; s >>= 1) {
        if (tid < s) red[tid] += red[tid + s];
        __syncthreads();
    }
    if (tid == 0) {
        const float shape_part = Rm[NSEQ * STRIDE + NSEQ];       // softDTW value
        const float cs = ALPHA_ / (float)NB;
        const float ct = (1.0f - ALPHA_) / ((float)NB * (float)NSEQ * (float)NSEQ);
        partials[b] = cs * shape_part + ct * red[0];
    }
}

// Single wave32: sum the 64 per-batch partials with one V_WMMA_F32_16X16X4_F32.
// A = 64 values as 16x4, B = ones(4x16) => D[m,n] = rowsum_m for every n,
// so sum(all D entries) = 16 * sum(partials) independent of the exact layout.
__global__ void __launch_bounds__(32)
dilate_reduce_kernel(const float* __restrict__ p, float* __restrict__ out) {
    const int L  = threadIdx.x;
    const int m  = L & 15;
    const int kb = (L < 16) ? 0 : 2;
    v2f a; a.x = p[m * 4 + kb]; a.y = p[m * 4 + kb + 1];
    v2f bo; bo.x = 1.0f; bo.y = 1.0f;
    v8f c = {};
    c = __builtin_amdgcn_wmma_f32_16x16x4_f32(false, a, false, bo,
                                              (short)0, c, false, false);
    float local = c[0] + c[1] + c[2] + c[3] + c[4] + c[5] + c[6] + c[7];
    local += __shfl_xor(local, 1,  32);
    local += __shfl_xor(local, 2,  32);
    local += __shfl_xor(local, 4,  32);
    local += __shfl_xor(local, 8,  32);
    local += __shfl_xor(local, 16, 32);
    if (L == 0) out[0] = local * (1.0f / 16.0f);
}

extern "C" void kernel_launch(void* const* d_in, const int* in_sizes, int n_in,
                              void* d_out, int out_size, void* d_ws, size_t ws_size,
                              hipStream_t stream) {
    (void)in_sizes; (void)n_in; (void)out_size; (void)ws_size;
    const float* y_pred = (const float*)d_in[0];   // o
    const float* y_true = (const float*)d_in[1];   // t
    float* partials = (float*)d_ws;                // 64 floats
    float* out      = (float*)d_out;               // 1 float

    dilate_batch_kernel<<<NB, 256, 0, stream>>>(y_pred, y_true, partials);
    dilate_reduce_kernel<<<1, 32, 0, stream>>>(partials, out);
}